// MDGCRNCell_13254269075713
// MI455X (gfx1250) — compile-verified
//
#include <hip/hip_runtime.h>
#include <math.h>

// MDGCRN cell for MI455X (gfx1250, wave32, WMMA).
// ws layout (floats): A[4194304] | bufX[16777216] | bufAX[16777216] | zr[16777216] | Wall[67108864]
// total ~465 MiB.

#define NN   2048
#define BB   64
#define CINC 128
#define DD   16
#define HID  64

typedef __attribute__((ext_vector_type(2))) float v2f;
typedef __attribute__((ext_vector_type(8))) float v8f;

__device__ __forceinline__ v8f wmma_f32(v2f a, v2f b, v8f c) {
  // V_WMMA_F32_16X16X4_F32 : D = A(16x4) * B(4x16) + C(16x16)
  return __builtin_amdgcn_wmma_f32_16x16x4_f32(false, a, false, b, (short)0, c, false, false);
}

__device__ __forceinline__ v8f vzero8() {
  v8f v;
#pragma unroll
  for (int i = 0; i < 8; ++i) v[i] = 0.0f;
  return v;
}

// ---------------- Kernel 1: A = softmax(relu(E E^T), axis=1) -----------------
__global__ __launch_bounds__(256) void build_A(const float* __restrict__ E,
                                               float* __restrict__ A) {
  const int n = blockIdx.x;
  const int t = threadIdx.x;
  __shared__ float sE[DD];
  __shared__ float red[256];
  if (t < DD) sE[t] = E[n * DD + t];
  __syncthreads();

  float sv[8];
  float lmax = -3.4e38f;
#pragma unroll
  for (int j = 0; j < 8; ++j) {
    const int m = t + j * 256;
    float s = 0.0f;
#pragma unroll
    for (int d = 0; d < DD; ++d) s = fmaf(sE[d], E[m * DD + d], s);
    s = fmaxf(s, 0.0f);
    sv[j] = s;
    lmax = fmaxf(lmax, s);
  }
  red[t] = lmax;
  __syncthreads();
  for (int s2 = 128; s2 > 0; s2 >>= 1) {
    if (t < s2) red[t] = fmaxf(red[t], red[t + s2]);
    __syncthreads();
  }
  const float gmax = red[0];
  __syncthreads();
  float lsum = 0.0f;
#pragma unroll
  for (int j = 0; j < 8; ++j) {
    sv[j] = __expf(sv[j] - gmax);
    lsum += sv[j];
  }
  red[t] = lsum;
  __syncthreads();
  for (int s2 = 128; s2 > 0; s2 >>= 1) {
    if (t < s2) red[t] += red[t + s2];
    __syncthreads();
  }
  const float inv = 1.0f / red[0];
#pragma unroll
  for (int j = 0; j < 8; ++j) A[(size_t)n * NN + t + j * 256] = sv[j] * inv;
}

// ------------- Kernel 2: build concat input (gate or candidate) --------------
// mode 0: [x, state, emb_dyn]   mode 1: [x, z*state, emb_dyn]  (z = zr[...,:64])
__global__ __launch_bounds__(256) void build_input(const float* __restrict__ x,
                                                   const float* __restrict__ state,
                                                   const float* __restrict__ ed,
                                                   const float* __restrict__ zr,
                                                   float* __restrict__ out, int mode) {
  const size_t idx = (size_t)blockIdx.x * 256 + threadIdx.x;
  const int c = (int)(idx & 127);
  const size_t bn = idx >> 7;  // b*N + n
  float v;
  if (c < 32) {
    v = x[bn * 32 + c];
  } else if (c < 96) {
    float s = state[bn * 64 + (c - 32)];
    if (mode) s *= zr[bn * 128 + (c - 32)];
    v = s;
  } else {
    v = ed[bn * 32 + (c - 96)];
  }
  out[idx] = v;
}

// ----------- Kernel 3: per-batch Y[b] = A[2048x2048] @ X[b][2048x128] --------
#define AK 32
__global__ __launch_bounds__(256) void gemm_AX(const float* __restrict__ A,
                                               const float* __restrict__ X,
                                               float* __restrict__ Y) {
  const int rowbase = blockIdx.x * 64;
  const int b = blockIdx.y;
  const int t = threadIdx.x;
  const int lane = t & 31, w = t >> 5;
  const int wm = w & 3, wn = w >> 2;         // 4 row-tiles x 2 col-halves
  const int lm = lane & 15, lk = (lane >> 4) * 2;

  __shared__ float sA[64 * 36];              // 64 rows x 32 k, padded stride 36
  __shared__ float sX[32 * 132];             // 32 k  x 128 n, padded stride 132

  const float* __restrict__ Xb = X + (size_t)b * NN * CINC;
  v8f acc[4];
#pragma unroll
  for (int i = 0; i < 4; ++i) acc[i] = vzero8();

  for (int kk = 0; kk < NN; kk += AK) {
    __syncthreads();
#pragma unroll
    for (int i = 0; i < 2; ++i) {            // A tile: 512 float4
      const int idx = t + i * 256;
      const int r = idx >> 3, k4 = (idx & 7) << 2;
      const float4 v = *(const float4*)&A[(size_t)(rowbase + r) * NN + kk + k4];
      *(float4*)&sA[r * 36 + k4] = v;
    }
#pragma unroll
    for (int i = 0; i < 4; ++i) {            // X tile: 1024 float4
      const int idx = t + i * 256;
      const int r = idx >> 5, c4 = (idx & 31) << 2;
      const float4 v = *(const float4*)&Xb[(size_t)(kk + r) * CINC + c4];
      *(float4*)&sX[r * 132 + c4] = v;
    }
    if (kk + AK < NN) {                      // global_prefetch_b8 of next A tile
      __builtin_prefetch(&A[(size_t)(rowbase + (t >> 3)) * NN + kk + AK], 0, 0);
    }
    __syncthreads();
#pragma unroll
    for (int k0 = 0; k0 < AK; k0 += 4) {
      const v2f a = *(const v2f*)&sA[(wm * 16 + lm) * 36 + k0 + lk];
#pragma unroll
      for (int tt = 0; tt < 4; ++tt) {
        v2f bf;
        bf.x = sX[(k0 + lk) * 132 + wn * 64 + tt * 16 + lm];
        bf.y = sX[(k0 + lk + 1) * 132 + wn * 64 + tt * 16 + lm];
        acc[tt] = wmma_f32(a, bf, acc[tt]);
      }
    }
  }
#pragma unroll
  for (int tt = 0; tt < 4; ++tt) {
#pragma unroll
    for (int r = 0; r < 8; ++r) {
      const int row = rowbase + wm * 16 + r + 8 * (lane >> 4);
      const int col = wn * 64 + tt * 16 + lm;
      Y[((size_t)b * NN + row) * CINC + col] = acc[tt][r];
    }
  }
}

// -------- Kernel 4: W_all[n, j] = sum_d E[n,d] * Wp[d, j], j in [0,KIO) ------
template <int KIO>
__global__ __launch_bounds__(256) void gemm_W(const float* __restrict__ E,
                                              const float* __restrict__ Wp,
                                              float* __restrict__ Wall) {
  const int t = threadIdx.x;
  const int jb = blockIdx.x, nb = blockIdx.y;
  __shared__ float sE[16 * DD];
  sE[t] = E[(size_t)nb * 16 * DD + t];       // 16 nodes x 16 dims == 256
  __syncthreads();
  const int j = jb * 256 + t;
  float acc[16];
#pragma unroll
  for (int i = 0; i < 16; ++i) acc[i] = 0.0f;
  for (int d = 0; d < DD; ++d) {
    const float wv = Wp[(size_t)d * KIO + j];
#pragma unroll
    for (int i = 0; i < 16; ++i) acc[i] = fmaf(sE[i * DD + d], wv, acc[i]);
  }
#pragma unroll
  for (int i = 0; i < 16; ++i) Wall[(size_t)(nb * 16 + i) * KIO + j] = acc[i];
}

// -------- Kernel 5: per-node GEMM [64 x 256] @ [256 x OUTC] + epilogue -------
// gate:   out = zr = sigmoid(Y + bias_n)              (OUTC=128)
// update: out = r*state + (1-r)*tanh(Y + bias_n)      (OUTC=64, r from zr)
template <int OUTC, bool IS_GATE>
__global__ __launch_bounds__(256) void node_out(const float* __restrict__ X0,
                                                const float* __restrict__ X1,
                                                const float* __restrict__ Wall,
                                                const float* __restrict__ E,
                                                const float* __restrict__ bp,
                                                const float* __restrict__ zr,
                                                const float* __restrict__ state,
                                                float* __restrict__ out) {
  const int n = blockIdx.x;
  const int t = threadIdx.x;
  const int lane = t & 31, w = t >> 5;
  const int wm = w & 3, wn = w >> 2;          // wn in {0,1}
  const int lm = lane & 15, lk = (lane >> 4) * 2;
  constexpr int NT = OUTC / 32;               // col tiles per wave
  constexpr int WS = OUTC + 4;                // padded W stride

  __shared__ float sXg[64 * 68];              // 64 b-rows x 64 k, stride 68
  __shared__ float sW[64 * WS];               // 64 k x OUTC
  __shared__ float sbias[OUTC];

  if (t < OUTC) {
    float s = 0.0f;
#pragma unroll
    for (int d = 0; d < DD; ++d) s = fmaf(E[n * DD + d], bp[d * OUTC + t], s);
    sbias[t] = s;
  }

  v8f acc[NT];
#pragma unroll
  for (int i = 0; i < NT; ++i) acc[i] = vzero8();

  const float* __restrict__ Wn = Wall + (size_t)n * 256 * OUTC;

  for (int kc = 0; kc < 4; ++kc) {
    __syncthreads();
    // Xg chunk: i = kc*64 .. +63 ; i<128 -> X0 (k=0 support), else X1 (A@X)
    const float* __restrict__ src =
        (kc < 2) ? (X0 + (size_t)n * CINC + (size_t)kc * 64)
                 : (X1 + (size_t)n * CINC + (size_t)(kc - 2) * 64);
#pragma unroll
    for (int i = 0; i < 16; ++i) {
      const int idx = t + i * 256;
      const int br = idx >> 6, ii = idx & 63;
      sXg[br * 68 + ii] = src[(size_t)br * NN * CINC + ii];
    }
    constexpr int F4PT = (64 * OUTC / 4) / 256;
#pragma unroll
    for (int i = 0; i < F4PT; ++i) {
      const int idx = t + i * 256;
      const int r = idx / (OUTC / 4), c4 = (idx % (OUTC / 4)) * 4;
      const float4 v = *(const float4*)&Wn[(size_t)(kc * 64 + r) * OUTC + c4];
      *(float4*)&sW[r * WS + c4] = v;
    }
    __syncthreads();
#pragma unroll
    for (int k0 = 0; k0 < 64; k0 += 4) {
      const v2f a = *(const v2f*)&sXg[(wm * 16 + lm) * 68 + k0 + lk];
#pragma unroll
      for (int tt = 0; tt < NT; ++tt) {
        const int col = wn * (OUTC / 2) + tt * 16 + lm;
        v2f bf;
        bf.x = sW[(k0 + lk) * WS + col];
        bf.y = sW[(k0 + lk + 1) * WS + col];
        acc[tt] = wmma_f32(a, bf, acc[tt]);
      }
    }
  }

#pragma unroll
  for (int tt = 0; tt < NT; ++tt) {
    const int col = wn * (OUTC / 2) + tt * 16 + lm;
#pragma unroll
    for (int r = 0; r < 8; ++r) {
      const int br = wm * 16 + r + 8 * (lane >> 4);     // batch index
      const float v = acc[tt][r] + sbias[col];
      const size_t bn = (size_t)br * NN + n;
      if constexpr (IS_GATE) {
        out[bn * 128 + col] = 1.0f / (1.0f + __expf(-v));
      } else {
        const float hc = tanhf(v);
        const float rr = zr[bn * 128 + 64 + col];
        const float st = state[bn * HID + col];
        out[bn * HID + col] = rr * st + (1.0f - rr) * hc;
      }
    }
  }
}

// ---------------------------------------------------------------------------
extern "C" void kernel_launch(void* const* d_in, const int* in_sizes, int n_in,
                              void* d_out, int out_size, void* d_ws, size_t ws_size,
                              hipStream_t stream) {
  const float* x  = (const float*)d_in[0];   // [B,N,32]
  const float* st = (const float*)d_in[1];   // [B,N,64]
  const float* ed = (const float*)d_in[2];   // [B,N,32]
  const float* E  = (const float*)d_in[3];   // [N,16]
  const float* Wg = (const float*)d_in[4];   // [16,2,128,128] -> [16,32768]
  const float* bg = (const float*)d_in[5];   // [16,128]
  const float* Wu = (const float*)d_in[6];   // [16,2,128,64]  -> [16,16384]
  const float* bu = (const float*)d_in[7];   // [16,64]
  float* out = (float*)d_out;                // [B,N,64]

  float* ws    = (float*)d_ws;
  float* A     = ws;                          // 4,194,304
  float* bufX  = A + (size_t)NN * NN;         // 16,777,216
  float* bufAX = bufX + (size_t)BB * NN * CINC;
  float* zr    = bufAX + (size_t)BB * NN * CINC;
  float* Wall  = zr + (size_t)BB * NN * 128;  // 67,108,864 (gate), reused for update

  const dim3 b256(256);
  const int ELEM_BLKS = (BB * NN * CINC) / 256;   // 65536

  // gate DGCC
  build_A<<<dim3(NN), b256, 0, stream>>>(E, A);
  build_input<<<dim3(ELEM_BLKS), b256, 0, stream>>>(x, st, ed, zr, bufX, 0);
  gemm_AX<<<dim3(NN / 64, BB), b256, 0, stream>>>(A, bufX, bufAX);
  gemm_W<32768><<<dim3(128, NN / 16), b256, 0, stream>>>(E, Wg, Wall);
  node_out<128, true><<<dim3(NN), b256, 0, stream>>>(bufX, bufAX, Wall, E, bg,
                                                     nullptr, nullptr, zr);
  // candidate DGCC + GRU combine
  build_input<<<dim3(ELEM_BLKS), b256, 0, stream>>>(x, st, ed, zr, bufX, 1);
  gemm_AX<<<dim3(NN / 64, BB), b256, 0, stream>>>(A, bufX, bufAX);
  gemm_W<16384><<<dim3(64, NN / 16), b256, 0, stream>>>(E, Wu, Wall);
  node_out<64, false><<<dim3(NN), b256, 0, stream>>>(bufX, bufAX, Wall, E, bu,
                                                     zr, st, out);
}